// SAGELayer_3332894622172
// MI455X (gfx1250) — compile-verified
//
#include <hip/hip_runtime.h>

typedef float v2f __attribute__((ext_vector_type(2)));
typedef float v8f __attribute__((ext_vector_type(8)));

#define DFEAT 256

// ---------------------------------------------------------------------------
// Kernel 1: zero the workspace accumulators (agg[N*256] ++ cnt[N]).
// ---------------------------------------------------------------------------
__global__ void sage_zero_f32(float* __restrict__ p, long long n) {
  long long i = (long long)blockIdx.x * blockDim.x + threadIdx.x;
  if (i < n) p[i] = 0.0f;
}

// ---------------------------------------------------------------------------
// Kernel 2: edge aggregation. One thread per (edge, 4-float chunk).
// x is 51 MB -> L2 resident on a 192 MB-L2 part; the gather (1.64 GB) and the
// 410M f32 atomics all stay in L2. float4 load -> 4 global_atomic_add_f32.
// ---------------------------------------------------------------------------
__global__ void __launch_bounds__(256)
sage_edge_agg(const float* __restrict__ x,
              const long long* __restrict__ ei,   // [2, E] int64: [0]=dst [1]=nbr
              float* __restrict__ agg,
              float* __restrict__ cnt,
              long long E) {
  long long tid = (long long)blockIdx.x * blockDim.x + threadIdx.x;
  long long total = E << 6;                        // 64 float4-chunks per edge
  if (tid >= total) return;
  long long e = tid >> 6;
  int c = ((int)(tid & 63)) << 2;                  // feature offset 0..252
  long long dst = ei[e];
  long long nbr = ei[E + e];
  const float4 v = *reinterpret_cast<const float4*>(x + nbr * DFEAT + c);
  float* a = agg + dst * DFEAT + c;
  atomicAdd(a + 0, v.x);
  atomicAdd(a + 1, v.y);
  atomicAdd(a + 2, v.z);
  atomicAdd(a + 3, v.w);
  if (c == 0) atomicAdd(cnt + dst, 1.0f);
}

// ---------------------------------------------------------------------------
// Kernel 3: out = x @ W1^T + b1 + (agg/max(cnt,1)) @ W2^T + b2
// One wave32 computes a 16x64 output tile with V_WMMA_F32_16X16X4_F32:
//   4 accumulators (v8f), A fragment reused across the 4 N-subtiles,
//   two K=256 passes (x/W1 then mean/W2) into the same accumulators.
// f32 WMMA fragment layout (ISA 7.12.2):
//   A: lane l<16 -> row l,  K = k0 + {0,1};  lane l>=16 -> row l-16, K = k0 + {2,3}
//   B: lane l<16 -> col l,  same K split (W1 row-major == B^T, so identical idx)
//   C/D: vgpr r -> M = r + 8*(lane/16), N = lane%16
// ---------------------------------------------------------------------------
__global__ void __launch_bounds__(256)
sage_wmma_gemm(const float* __restrict__ x,
               const float* __restrict__ W1,
               const float* __restrict__ bias1,
               const float* __restrict__ W2,
               const float* __restrict__ bias2,
               const float* __restrict__ agg,
               const float* __restrict__ cnt,
               float* __restrict__ out,
               int N) {
  const int wid  = blockIdx.x * (blockDim.x >> 5) + (threadIdx.x >> 5);
  const int lane = threadIdx.x & 31;
  const int tm = wid >> 2;            // DFEAT/64 == 4 column tiles per row strip
  const int tn = wid & 3;
  const int m0 = tm << 4;
  if (m0 >= N) return;
  const int n0 = tn << 6;

  const int half = lane >> 4;         // 0 or 1
  const int lr   = lane & 15;         // A-row / B-col within 16
  const int koff = half << 1;         // K sub-offset {0,2}

  const int  arow  = m0 + lr;
  const bool rowOK = arow < N;
  const long long abase = (long long)(rowOK ? arow : 0) * DFEAT;
  const float* xrow = x   + abase;
  const float* grow = agg + abase;
  const float  invc = rowOK ? (1.0f / fmaxf(cnt[arow], 1.0f)) : 0.0f;

  const long long wrow0 = (long long)(n0 +  0 + lr) * DFEAT;
  const long long wrow1 = (long long)(n0 + 16 + lr) * DFEAT;
  const long long wrow2 = (long long)(n0 + 32 + lr) * DFEAT;
  const long long wrow3 = (long long)(n0 + 48 + lr) * DFEAT;

  v8f acc[4] = {v8f{}, v8f{}, v8f{}, v8f{}};

  // Pass 1: x @ W1^T
  for (int k0 = 0; k0 < DFEAT; k0 += 4) {
    const v2f a  = *reinterpret_cast<const v2f*>(xrow + k0 + koff);
    const v2f b0 = *reinterpret_cast<const v2f*>(W1 + wrow0 + k0 + koff);
    const v2f b1 = *reinterpret_cast<const v2f*>(W1 + wrow1 + k0 + koff);
    const v2f b2 = *reinterpret_cast<const v2f*>(W1 + wrow2 + k0 + koff);
    const v2f b3 = *reinterpret_cast<const v2f*>(W1 + wrow3 + k0 + koff);
    acc[0] = __builtin_amdgcn_wmma_f32_16x16x4_f32(false, a, false, b0, (short)0, acc[0], false, false);
    acc[1] = __builtin_amdgcn_wmma_f32_16x16x4_f32(false, a, false, b1, (short)0, acc[1], false, false);
    acc[2] = __builtin_amdgcn_wmma_f32_16x16x4_f32(false, a, false, b2, (short)0, acc[2], false, false);
    acc[3] = __builtin_amdgcn_wmma_f32_16x16x4_f32(false, a, false, b3, (short)0, acc[3], false, false);
  }

  // Pass 2: (agg * invc) @ W2^T  (mean division fused into A load)
  for (int k0 = 0; k0 < DFEAT; k0 += 4) {
    v2f a = *reinterpret_cast<const v2f*>(grow + k0 + koff);
    a.x *= invc;
    a.y *= invc;
    const v2f b0 = *reinterpret_cast<const v2f*>(W2 + wrow0 + k0 + koff);
    const v2f b1 = *reinterpret_cast<const v2f*>(W2 + wrow1 + k0 + koff);
    const v2f b2 = *reinterpret_cast<const v2f*>(W2 + wrow2 + k0 + koff);
    const v2f b3 = *reinterpret_cast<const v2f*>(W2 + wrow3 + k0 + koff);
    acc[0] = __builtin_amdgcn_wmma_f32_16x16x4_f32(false, a, false, b0, (short)0, acc[0], false, false);
    acc[1] = __builtin_amdgcn_wmma_f32_16x16x4_f32(false, a, false, b1, (short)0, acc[1], false, false);
    acc[2] = __builtin_amdgcn_wmma_f32_16x16x4_f32(false, a, false, b2, (short)0, acc[2], false, false);
    acc[3] = __builtin_amdgcn_wmma_f32_16x16x4_f32(false, a, false, b3, (short)0, acc[3], false, false);
  }

  // Bias + store. D layout: vgpr r -> row m0 + r + 8*half, col n0 + 16*j + lr.
#pragma unroll
  for (int j = 0; j < 4; ++j) {
    const int col = n0 + (j << 4) + lr;
    const float bb = bias1[col] + bias2[col];
#pragma unroll
    for (int r = 0; r < 8; ++r) {
      const int orow = m0 + r + (half << 3);
      if (orow < N)
        out[(long long)orow * DFEAT + col] = acc[j][r] + bb;
    }
  }
}

// ---------------------------------------------------------------------------
extern "C" void kernel_launch(void* const* d_in, const int* in_sizes, int n_in,
                              void* d_out, int out_size, void* d_ws, size_t ws_size,
                              hipStream_t stream) {
  const float*     x   = (const float*)d_in[0];
  const float*     W1  = (const float*)d_in[1];
  const float*     b1  = (const float*)d_in[2];
  const float*     W2  = (const float*)d_in[3];
  const float*     b2  = (const float*)d_in[4];
  const long long* ei  = (const long long*)d_in[5];   // int64 [2, E]

  const long long N = in_sizes[0] / DFEAT;
  const long long E = in_sizes[5] / 2;

  float* agg = (float*)d_ws;          // N * 256
  float* cnt = agg + N * DFEAT;       // N
  float* out = (float*)d_out;

  {
    long long n = N * DFEAT + N;
    unsigned blocks = (unsigned)((n + 255) / 256);
    sage_zero_f32<<<blocks, 256, 0, stream>>>(agg, n);
  }
  {
    long long total = E << 6;         // E * 64 chunk-threads
    unsigned blocks = (unsigned)((total + 255) / 256);
    sage_edge_agg<<<blocks, 256, 0, stream>>>(x, ei, agg, cnt, E);
  }
  {
    long long waves  = ((N + 15) / 16) * 4;   // 16x64 tile per wave
    unsigned blocks = (unsigned)((waves + 7) / 8);  // 8 waves / 256-thread block
    sage_wmma_gemm<<<blocks, 256, 0, stream>>>(x, W1, b1, W2, b2, agg, cnt, out, (int)N);
  }
}